// Boat_Dynamic_System_13872744366207
// MI455X (gfx1250) — compile-verified
//
#include <hip/hip_runtime.h>

// CDNA5 / gfx1250, wave32. FP32 WMMA path: V_WMMA_F32_16X16X4_F32.
typedef __attribute__((ext_vector_type(2))) float v2f;
typedef __attribute__((ext_vector_type(8))) float v8f;

#define TPW 4  // tiles (of 16 rows) per wave, batched loads for MLP

__global__ __launch_bounds__(256) void boat_wmma_kernel(
    const float* __restrict__ tptr,    // (1,)
    const float* __restrict__ state,   // (B,4) row-major
    const float* __restrict__ cmd,     // (T,2)
    const float* __restrict__ coeffs,  // (4,45)
    float* __restrict__ out,           // (B,4)
    int nrows, int ntiles)
{
    const int lane   = threadIdx.x & 31;
    const int waveId = (blockIdx.x * blockDim.x + threadIdx.x) >> 5;

    // ---- scalar command select: idx = round(t/DT), round-to-nearest-even ----
    const float t0  = tptr[0];
    const int   idx = __float2int_rn(t0 * 100.0f);
    const float pro = cmd[2 * idx + 0];
    const float rud = cmd[2 * idx + 1];

    // ---- A matrix = folded coefficients Ceff (4x15 in a 16x16xK=4-chunk frame).
    // 32-bit A 16x4 layout: VGPR0 = K=4g (lanes 0-15) / K=4g+2 (lanes 16-31),
    // VGPR1 = K+1. M = lane%16 = output column c (only c<4 live).
    const int  cc = lane & 15;
    const bool hi = lane >= 16;

    v2f A[4];
#pragma unroll
    for (int g = 0; g < 4; ++g) {
        const int j0 = 4 * g + (hi ? 2 : 0);
        float a0 = 0.0f, a1 = 0.0f;
        if (cc < 4) {
            const float* cr = coeffs + cc * 45;
            if (j0 < 15)     a0 = cr[j0]     + pro * cr[15 + j0]     + rud * cr[30 + j0];
            if (j0 + 1 < 15) a1 = cr[j0 + 1] + pro * cr[15 + j0 + 1] + rud * cr[30 + j0 + 1];
        }
        A[g][0] = a0;
        A[g][1] = a1;
    }

    const float4* state4 = reinterpret_cast<const float4*>(state);
    float4*       out4   = reinterpret_cast<float4*>(out);

    const int tbase = waveId * TPW;

    // ---- batched, coalesced row loads: lane L fetches state row (tile*16 + L%16).
    // High lanes hit the same two 128B lines as low lanes -> no extra traffic.
    bool   valid[TPW];
    float4 s[TPW];
#pragma unroll
    for (int k = 0; k < TPW; ++k) {
        const int tile = tbase + k;
        valid[k] = (tile < ntiles);            // wave-uniform
        const int row = tile * 16 + cc;
        s[k] = make_float4(0.f, 0.f, 0.f, 0.f);
        if (valid[k] && row < nrows) s[k] = state4[row];
    }

#pragma unroll
    for (int k = 0; k < TPW; ++k) {
        if (!valid[k]) continue;               // wave-uniform: EXEC full at WMMA
        const float u = s[k].x, v = s[k].y, r = s[k].z, P = s[k].w;
        // 15 quadratic features + zero pad to K=16
        float f[16] = {1.0f, u, v, r, P,
                       u * u, u * v, u * r, u * P,
                       v * v, v * r, v * P,
                       r * r, r * P, P * P, 0.0f};

        // B = base^T (KxN, N = tile row). Same K striping as A.
        v8f acc = {};
#pragma unroll
        for (int g = 0; g < 4; ++g) {
            v2f Bv;
            Bv[0] = hi ? f[4 * g + 2] : f[4 * g + 0];
            Bv[1] = hi ? f[4 * g + 3] : f[4 * g + 1];
            // D = A x B + C : 16x16 f32 accumulate over K chunks (4 chained)
            acc = __builtin_amdgcn_wmma_f32_16x16x4_f32(
                false, A[g], false, Bv, (short)0, acc, false, false);
        }

        // D layout: VGPR v, lanes 0-15 -> M=v (=c), N=lane (=row-in-tile).
        // So lane n owns out[b0+n][0..3] contiguously -> coalesced 256B store.
        const int tile = tbase + k;
        const int row  = tile * 16 + cc;
        if (lane < 16 && row < nrows) {
            float4 o = make_float4(acc[0], acc[1], acc[2], acc[3]);
            out4[row] = o;
        }
    }
}

extern "C" void kernel_launch(void* const* d_in, const int* in_sizes, int n_in,
                              void* d_out, int out_size, void* d_ws, size_t ws_size,
                              hipStream_t stream) {
    const float* t      = (const float*)d_in[0];
    const float* state  = (const float*)d_in[1];
    const float* cmd    = (const float*)d_in[2];
    const float* coeffs = (const float*)d_in[3];
    float* out = (float*)d_out;

    const int nrows  = in_sizes[1] / 4;                 // B = 2,097,152
    const int ntiles = (nrows + 15) / 16;               // 131072 16-row tiles
    const int waves  = (ntiles + TPW - 1) / TPW;        // 32768 waves
    const int block  = 256;                             // 8 waves / block
    const int wavesPerBlock = block / 32;
    const int grid   = (waves + wavesPerBlock - 1) / wavesPerBlock;  // 4096

    boat_wmma_kernel<<<grid, block, 0, stream>>>(t, state, cmd, coeffs, out,
                                                 nrows, ntiles);
}